// GraphAttention_4398046511968
// MI455X (gfx1250) — compile-verified
//
#include <hip/hip_runtime.h>
#include <math.h>

typedef __attribute__((ext_vector_type(2))) float v2f;
typedef __attribute__((ext_vector_type(8))) float v8f;

#define L_DIM 17
#define N_DIM 128
#define O_DIM 96
#define K_DIM 300
#define I_DIM 32
#define J_DIM 2
#define M_DIM (I_DIM * K_DIM)   /* 9600 */
#define BN_EPS 1e-3f
#define LDSP 98                  /* LDS pitch for transposed B: even (8B align), bank-friendly */

// ---------------------------------------------------------------------------
// Kernel 1: fold softmax-attention sum + BatchNorm into per-(l,n) scale and
// per-n bias.  s[l][n] = (sum_m softmax_n(mat[m,l]*amask[m,l,:]))[n] * inv[n]
// b[n] = beta[n] - mean[n]*inv[n],  inv = gamma * rsqrt(var+eps)
// ---------------------------------------------------------------------------
__global__ __launch_bounds__(N_DIM) void prep_kernel(
    const float* __restrict__ mat, const float* __restrict__ amask,
    const float* __restrict__ gamma, const float* __restrict__ beta,
    const float* __restrict__ mmean, const float* __restrict__ mvar,
    float* __restrict__ s_out, float* __restrict__ b_out) {
  __shared__ float red[N_DIM];
  const int l = blockIdx.x;
  const int n = threadIdx.x;
  float acc = 0.f;
  for (int m = 0; m < L_DIM; ++m) {
    const float v = mat[m * L_DIM + l] * amask[(m * L_DIM + l) * N_DIM + n];
    red[n] = v; __syncthreads();
    for (int st = N_DIM / 2; st > 0; st >>= 1) {
      if (n < st) red[n] = fmaxf(red[n], red[n + st]);
      __syncthreads();
    }
    const float mx = red[0]; __syncthreads();
    const float e = __expf(v - mx);
    red[n] = e; __syncthreads();
    for (int st = N_DIM / 2; st > 0; st >>= 1) {
      if (n < st) red[n] += red[n + st];
      __syncthreads();
    }
    const float sum = red[0]; __syncthreads();
    acc += e / sum;
  }
  const float inv = gamma[n] * __frsqrt_rn(mvar[n] + BN_EPS);
  s_out[l * N_DIM + n] = acc * inv;
  if (l == 0) b_out[n] = beta[n] - mmean[n] * inv;
}

// ---------------------------------------------------------------------------
// Kernel 2: per-l GEMM (9600x96 @ 96x128) with f32 WMMA, fused j-sum of x,
// fused scale/bias/ReLU epilogue.  Grid (75, 17), 256 threads = 8 waves.
// Each wave: one 16-row strip, all 128 columns (8 accumulator tiles).
// ---------------------------------------------------------------------------
__global__ __launch_bounds__(256) void gemm_kernel(
    const float* __restrict__ x, const float* __restrict__ kern,
    const float* __restrict__ s_in, const float* __restrict__ b_in,
    float* __restrict__ out) {
  __shared__ float Bs[N_DIM * LDSP];   // kernel[l] transposed: Bs[n*LDSP + o]

  const int l   = blockIdx.y;
  const int tid = threadIdx.x;

  // Stage kernel[l] (96x128, row-major over n) into LDS transposed [n][o].
  for (int idx = tid; idx < O_DIM * N_DIM; idx += 256) {
    const int o = idx >> 7;       // idx / 128
    const int n = idx & 127;      // idx % 128  (coalesced over n)
    Bs[n * LDSP + o] = kern[((size_t)l * O_DIM + o) * N_DIM + n];
  }
  __syncthreads();

  const int wave = tid >> 5;
  const int lane = tid & 31;
  const int ln   = lane & 15;     // M (for A) / N (for B,C,D) within tile
  const int h    = lane >> 4;     // lane-half: K offset 2h (A,B), row offset 8h (C,D)

  const int mtile = blockIdx.x * 8 + wave;      // 0..599
  const int m = mtile * 16 + ln;                // A-row this lane feeds
  const int i = m / K_DIM;
  const int k = m - i * K_DIM;
  // x[i, j, k, l, o]: o stride 1, l stride 96, k stride 17*96, j stride 300*17*96
  const float* pA = x + ((size_t)i * J_DIM * K_DIM + k) * (L_DIM * O_DIM)
                      + (size_t)l * O_DIM + 2 * h;
  const size_t jstr = (size_t)K_DIM * L_DIM * O_DIM;   // 489600

  const v8f zero = {0.f, 0.f, 0.f, 0.f, 0.f, 0.f, 0.f, 0.f};
  v8f acc[8];
  #pragma unroll
  for (int t = 0; t < 8; ++t) acc[t] = zero;

  const int ob0 = 2 * h;
  #pragma unroll 4
  for (int ko = 0; ko < O_DIM; ko += 4) {
    // A fragment: lane holds {A[m][ko+2h], A[m][ko+2h+1]}, j-summed on the fly
    const float2 a0 = *(const float2*)(pA + ko);
    const float2 a1 = *(const float2*)(pA + jstr + ko);
    v2f a; a.x = a0.x + a1.x; a.y = a0.y + a1.y;
    const int ob = ko + ob0;
    #pragma unroll
    for (int nt = 0; nt < 8; ++nt) {
      // B fragment: lane holds {B[ko+2h][n], B[ko+2h+1][n]} -> contiguous in Bs
      const float2 bv = *(const float2*)(&Bs[(nt * 16 + ln) * LDSP + ob]);
      v2f b; b.x = bv.x; b.y = bv.y;
      acc[nt] = __builtin_amdgcn_wmma_f32_16x16x4_f32(
          false, a, false, b, (short)0, acc[nt], false, false);
    }
  }

  // Fused epilogue: out = relu(acc * s[l][n] + b[n])
  float sc[8], bb[8];
  #pragma unroll
  for (int nt = 0; nt < 8; ++nt) {
    sc[nt] = s_in[l * N_DIM + nt * 16 + ln];
    bb[nt] = b_in[nt * 16 + ln];
  }
  #pragma unroll
  for (int r = 0; r < 8; ++r) {
    const int mrow = mtile * 16 + r + 8 * h;   // C/D layout: VGPR r, half h
    const int ii = mrow / K_DIM;
    const int kk = mrow - ii * K_DIM;
    float* po = out + (((size_t)ii * K_DIM + kk) * L_DIM + l) * N_DIM;
    #pragma unroll
    for (int nt = 0; nt < 8; ++nt) {
      const float v = acc[nt][r] * sc[nt] + bb[nt];
      po[nt * 16 + ln] = fmaxf(v, 0.f);
    }
  }
}

// ---------------------------------------------------------------------------
extern "C" void kernel_launch(void* const* d_in, const int* in_sizes, int n_in,
                              void* d_out, int out_size, void* d_ws, size_t ws_size,
                              hipStream_t stream) {
  const float* x     = (const float*)d_in[0];
  const float* mat   = (const float*)d_in[1];
  const float* amask = (const float*)d_in[2];
  const float* kern  = (const float*)d_in[3];
  const float* gamma = (const float*)d_in[4];
  const float* beta  = (const float*)d_in[5];
  const float* mmean = (const float*)d_in[6];
  const float* mvar  = (const float*)d_in[7];
  float* outp = (float*)d_out;

  float* wsf = (float*)d_ws;
  float* s   = wsf;                       // [17*128]
  float* b   = wsf + L_DIM * N_DIM;       // [128]

  prep_kernel<<<dim3(L_DIM), dim3(N_DIM), 0, stream>>>(
      mat, amask, gamma, beta, mmean, mvar, s, b);

  gemm_kernel<<<dim3(M_DIM / (16 * 8), L_DIM), dim3(256), 0, stream>>>(
      x, kern, s, b, outp);
}